// GCN_v1_38620345926216
// MI455X (gfx1250) — compile-verified
//
#include <hip/hip_runtime.h>
#include <math.h>

typedef __attribute__((ext_vector_type(2))) float v2f;
typedef __attribute__((ext_vector_type(8))) float v8f;

namespace {

constexpr int kN    = 50000;
constexpr int kE    = 600000;
constexpr int kEtot = kE + kN;
constexpr float kNegSlope = 0.2f;

// Ordered-uint mapping so float max == uint max (handles negatives).
__device__ __forceinline__ unsigned f2ord(float f) {
  unsigned u = __float_as_uint(f);
  return (u & 0x80000000u) ? ~u : (u | 0x80000000u);
}
__device__ __forceinline__ float ord2f(unsigned u) {
  u = (u & 0x80000000u) ? (u & 0x7fffffffu) : ~u;
  return __uint_as_float(u);
}

// ---- graph preprocessing -------------------------------------------------
__global__ void k_zero_graph(float* deg, float* ea_loop /* kN*8 */) {
  int i = blockIdx.x * blockDim.x + threadIdx.x;
  if (i < kN * 8) ea_loop[i] = 0.f;
  if (i < kN) deg[i] = 0.f;
}

__global__ void k_build_edges(const long long* __restrict__ ei,
                              const float* __restrict__ ea,
                              int* __restrict__ src, int* __restrict__ dst,
                              float* __restrict__ ea_loop, float* __restrict__ deg) {
  int e = blockIdx.x * blockDim.x + threadIdx.x;
  if (e >= kE) return;
  int s = (int)ei[e];
  int d = (int)ei[(size_t)kE + e];
  src[e] = s;
  dst[e] = d;
  atomicAdd(&deg[d], 1.f);
  const float* row = ea + (size_t)e * 8;
  float* lrow = ea_loop + (size_t)d * 8;
#pragma unroll
  for (int k = 0; k < 8; ++k) atomicAdd(&lrow[k], row[k]);
}

__global__ void k_self_loops(int* __restrict__ src, int* __restrict__ dst,
                             float* __restrict__ ea_loop, const float* __restrict__ deg) {
  int n = blockIdx.x * blockDim.x + threadIdx.x;
  if (n >= kN) return;
  src[kE + n] = n;
  dst[kE + n] = n;
  float inv = 1.f / fmaxf(deg[n], 1.f);
  float* row = ea_loop + (size_t)n * 8;
#pragma unroll
  for (int k = 0; k < 8; ++k) row[k] *= inv;
}

// ---- per-layer init: zero accumulators, compute v = We @ ae (8 floats) ---
template <int C>
__global__ void k_layer_init(float* __restrict__ acc, unsigned* __restrict__ m_u,
                             float* __restrict__ denom, float* __restrict__ v,
                             const float* __restrict__ We, const float* __restrict__ ae) {
  int i = blockIdx.x * blockDim.x + threadIdx.x;
  if (i < kN * C) acc[i] = 0.f;
  if (i < kN) { m_u[i] = 0u; denom[i] = 0.f; }
  if (i < 8) {
    float s = 0.f;
    for (int c = 0; c < C; ++c) s += We[i * C + c] * ae[c];
    v[i] = s;
  }
}

// ---- H = X @ W via V_WMMA_F32_16X16X4_F32, one 16x16 tile per wave -------
// K fixed at 128. Layouts per CDNA5 ISA 7.12.2:
//   A 16x4 f32 : M=lane%16, K = 2*(lane/16) + vgpr
//   B 4x16 f32 : N=lane%16, K = 2*(lane/16) + vgpr
//   D 16x16    : N=lane%16, M = vgpr + 8*(lane/16)
template <int C>
__global__ void k_gemm_wmma(const float* __restrict__ X, const float* __restrict__ W,
                            float* __restrict__ H, int tiles_total) {
  constexpr int K = 128;
  constexpr int tiles_n = C / 16;
  int wave = threadIdx.x >> 5;
  int tile = blockIdx.x * (blockDim.x >> 5) + wave;
  if (tile >= tiles_total) return;  // wave-uniform: EXEC stays all-ones
  int tm = tile / tiles_n;
  int tn = tile - tm * tiles_n;
  int lane = threadIdx.x & 31;
  int m  = lane & 15;
  int kg = lane >> 4;
  int rowbase = tm * 16;
  int colbase = tn * 16;
  const float* xr = X + (size_t)(rowbase + m) * K + 2 * kg;
  v8f c = {};
#pragma unroll
  for (int k = 0; k < K; k += 4) {
    v2f a;
    a.x = xr[k];
    a.y = xr[k + 1];
    int kb = k + 2 * kg;
    v2f b;
    b.x = W[(size_t)kb * C + colbase + m];
    b.y = W[(size_t)(kb + 1) * C + colbase + m];
    c = __builtin_amdgcn_wmma_f32_16x16x4_f32(false, a, false, b, (short)0, c,
                                              false, false);
  }
#pragma unroll
  for (int r = 0; r < 8; ++r)
    H[(size_t)(rowbase + r + 8 * kg) * C + colbase + m] = c[r];
}

// ---- per-node row dots hs = H@a_s, hd = H@a_d (one wave per node) --------
template <int C>
__global__ void k_rowdot(const float* __restrict__ H, const float* __restrict__ as_,
                         const float* __restrict__ ad_, float* __restrict__ hs,
                         float* __restrict__ hd) {
  int wave = threadIdx.x >> 5;
  int lane = threadIdx.x & 31;
  int n = blockIdx.x * (blockDim.x >> 5) + wave;
  if (n >= kN) return;
  float s = 0.f, d = 0.f;
  const float* row = H + (size_t)n * C;
#pragma unroll
  for (int c = lane; c < C; c += 32) {
    float h = row[c];
    s += h * as_[c];
    d += h * ad_[c];
  }
#pragma unroll
  for (int off = 16; off > 0; off >>= 1) {
    s += __shfl_down(s, off);
    d += __shfl_down(d, off);
  }
  if (lane == 0) { hs[n] = s; hd[n] = d; }
}

// ---- per-edge attention logit + segment-max ------------------------------
__global__ void k_alpha(const int* __restrict__ src, const int* __restrict__ dst,
                        const float* __restrict__ ea, const float* __restrict__ ea_loop,
                        const float* __restrict__ hs, const float* __restrict__ hd,
                        const float* __restrict__ v, float* __restrict__ alpha,
                        unsigned* __restrict__ m_u) {
  int e = blockIdx.x * blockDim.x + threadIdx.x;
  if (e >= kEtot) return;
  const float* row = (e < kE) ? (ea + (size_t)e * 8)
                              : (ea_loop + (size_t)(e - kE) * 8);
  float t = 0.f;
#pragma unroll
  for (int k = 0; k < 8; ++k) t += row[k] * v[k];
  int d = dst[e];
  float a = hs[src[e]] + hd[d] + t;
  a = (a > 0.f) ? a : kNegSlope * a;
  alpha[e] = a;
  atomicMax(&m_u[d], f2ord(a));
}

// ---- exp(alpha - m[dst]) and segment-sum denominator ---------------------
__global__ void k_exp(const int* __restrict__ dst, const unsigned* __restrict__ m_u,
                      float* __restrict__ alpha, float* __restrict__ denom) {
  int e = blockIdx.x * blockDim.x + threadIdx.x;
  if (e >= kEtot) return;
  int d = dst[e];
  float ex = expf(alpha[e] - ord2f(m_u[d]));
  alpha[e] = ex;
  atomicAdd(&denom[d], ex);
}

// ---- weighted scatter-add of h[src] rows into acc[dst] -------------------
template <int C>
__global__ void k_scatter(const int* __restrict__ src, const int* __restrict__ dst,
                          const float* __restrict__ alpha, const float* __restrict__ denom,
                          const float* __restrict__ H, float* __restrict__ acc) {
  int e = blockIdx.x;
  int c = threadIdx.x;
  int s = src[e], d = dst[e];
  float w = alpha[e] / (denom[d] + 1e-16f);
  atomicAdd(&acc[(size_t)d * C + c], H[(size_t)s * C + c] * w);
}

// ---- bias + optional relu, in place --------------------------------------
template <int C>
__global__ void k_epilogue(float* __restrict__ acc, const float* __restrict__ bias,
                           int do_relu) {
  int i = blockIdx.x * blockDim.x + threadIdx.x;
  if (i >= kN * C) return;
  float y = acc[i] + bias[i & (C - 1)];
  if (do_relu) y = fmaxf(y, 0.f);
  acc[i] = y;
}

}  // namespace

extern "C" void kernel_launch(void* const* d_in, const int* in_sizes, int n_in,
                              void* d_out, int out_size, void* d_ws, size_t ws_size,
                              hipStream_t stream) {
  (void)in_sizes; (void)n_in; (void)out_size; (void)ws_size;
  const float*     x  = (const float*)d_in[0];
  const long long* ei = (const long long*)d_in[1];
  const float*     ea = (const float*)d_in[2];
  const float* W[3]  = {(const float*)d_in[3], (const float*)d_in[9],  (const float*)d_in[15]};
  const float* as_[3]= {(const float*)d_in[4], (const float*)d_in[10], (const float*)d_in[16]};
  const float* ad_[3]= {(const float*)d_in[5], (const float*)d_in[11], (const float*)d_in[17]};
  const float* We[3] = {(const float*)d_in[6], (const float*)d_in[12], (const float*)d_in[18]};
  const float* ae[3] = {(const float*)d_in[7], (const float*)d_in[13], (const float*)d_in[19]};
  const float* bb[3] = {(const float*)d_in[8], (const float*)d_in[14], (const float*)d_in[20]};

  char* ws = (char*)d_ws;
  size_t off = 0;
  auto alloc = [&](size_t bytes) -> void* {
    void* p = ws + off;
    off += (bytes + 255) & ~(size_t)255;
    return p;
  };
  int*      src     = (int*)     alloc(sizeof(int) * (size_t)kEtot);
  int*      dst     = (int*)     alloc(sizeof(int) * (size_t)kEtot);
  float*    ea_loop = (float*)   alloc(sizeof(float) * (size_t)kN * 8);
  float*    deg     = (float*)   alloc(sizeof(float) * (size_t)kN);
  float*    H       = (float*)   alloc(sizeof(float) * (size_t)kN * 128);
  float*    X1      = (float*)   alloc(sizeof(float) * (size_t)kN * 128);
  float*    X2      = (float*)   alloc(sizeof(float) * (size_t)kN * 128);
  float*    hs      = (float*)   alloc(sizeof(float) * (size_t)kN);
  float*    hd      = (float*)   alloc(sizeof(float) * (size_t)kN);
  float*    alpha   = (float*)   alloc(sizeof(float) * (size_t)kEtot);
  unsigned* m_u     = (unsigned*)alloc(sizeof(unsigned) * (size_t)kN);
  float*    denom   = (float*)   alloc(sizeof(float) * (size_t)kN);
  float*    vvec    = (float*)   alloc(sizeof(float) * 8);

  const int TB = 256;
  k_zero_graph <<<(kN * 8 + TB - 1) / TB, TB, 0, stream>>>(deg, ea_loop);
  k_build_edges<<<(kE + TB - 1) / TB, TB, 0, stream>>>(ei, ea, src, dst, ea_loop, deg);
  k_self_loops <<<(kN + TB - 1) / TB, TB, 0, stream>>>(src, dst, ea_loop, deg);

  const float* Xin = x;
  for (int layer = 0; layer < 3; ++layer) {
    const int C = (layer == 2) ? 64 : 128;
    float* acc = (layer == 0) ? X1 : (layer == 1) ? X2 : (float*)d_out;
    const int tiles_total = (kN / 16) * (C / 16);
    const int gNC = (kN * C + TB - 1) / TB;
    const int gE  = (kEtot + TB - 1) / TB;
    if (C == 128) {
      k_layer_init<128><<<gNC, TB, 0, stream>>>(acc, m_u, denom, vvec, We[layer], ae[layer]);
      k_gemm_wmma <128><<<(tiles_total + 7) / 8, 256, 0, stream>>>(Xin, W[layer], H, tiles_total);
      k_rowdot    <128><<<(kN + 3) / 4, 128, 0, stream>>>(H, as_[layer], ad_[layer], hs, hd);
    } else {
      k_layer_init<64><<<gNC, TB, 0, stream>>>(acc, m_u, denom, vvec, We[layer], ae[layer]);
      k_gemm_wmma <64><<<(tiles_total + 7) / 8, 256, 0, stream>>>(Xin, W[layer], H, tiles_total);
      k_rowdot    <64><<<(kN + 3) / 4, 128, 0, stream>>>(H, as_[layer], ad_[layer], hs, hd);
    }
    k_alpha<<<gE, TB, 0, stream>>>(src, dst, ea, ea_loop, hs, hd, vvec, alpha, m_u);
    k_exp  <<<gE, TB, 0, stream>>>(dst, m_u, alpha, denom);
    if (C == 128) {
      k_scatter <128><<<kEtot, 128, 0, stream>>>(src, dst, alpha, denom, H, acc);
      k_epilogue<128><<<gNC, TB, 0, stream>>>(acc, bb[layer], 1);
    } else {
      k_scatter <64><<<kEtot, 64, 0, stream>>>(src, dst, alpha, denom, H, acc);
      k_epilogue<64><<<gNC, TB, 0, stream>>>(acc, bb[layer], 0);
    }
    Xin = acc;
  }
}